// HANModel_24739011625781
// MI455X (gfx1250) — compile-verified
//
#include <hip/hip_runtime.h>
#include <math.h>

#define HIDC 128
#define HEADS 8
#define DH 16
#define NEG_SLOPE 0.2f

typedef __attribute__((ext_vector_type(2))) float v2f;
typedef __attribute__((ext_vector_type(8))) float v8f;

// ---------- ordered-uint encoding for float atomic max ----------
__device__ __forceinline__ unsigned ord_enc(float x) {
  unsigned u = __float_as_uint(x);
  return (u & 0x80000000u) ? ~u : (u | 0x80000000u);
}
__device__ __forceinline__ float ord_dec(unsigned e) {
  unsigned u = (e & 0x80000000u) ? (e & 0x7fffffffu) : ~e;
  return __uint_as_float(u);
}
#define ORD_NEG_INF 0x007FFFFFu  // ord_enc(-inf)

// ---------- small utility kernels ----------
__global__ void k_zero_f32(float* __restrict__ p, long n) {
  long i = (long)blockIdx.x * blockDim.x + threadIdx.x;
  if (i < n) p[i] = 0.f;
}
__global__ void k_fill_u32(unsigned* __restrict__ p, long n, unsigned v) {
  long i = (long)blockIdx.x * blockDim.x + threadIdx.x;
  if (i < n) p[i] = v;
}
__global__ void k_transpose128(const float* __restrict__ W, float* __restrict__ Wt) {
  int i = blockIdx.x * blockDim.x + threadIdx.x;   // 16384 threads
  int k = i >> 7, n = i & 127;
  Wt[n * HIDC + k] = W[k * HIDC + n];
}
__global__ void k_relu(float* __restrict__ p, long n) {
  long i = (long)blockIdx.x * blockDim.x + threadIdx.x;
  if (i < n) p[i] = fmaxf(p[i], 0.f);
}
__global__ void k_elu_inplace(float* __restrict__ p, long n) {
  long i = (long)blockIdx.x * blockDim.x + threadIdx.x;
  if (i < n) { float v = p[i]; p[i] = v > 0.f ? v : expm1f(v); }
}

// ---------- fp32 WMMA GEMM: C[N,128] = A[N,128] @ W[128,128] + bias ----------
// Bt is W transposed (Bt[n*128+k] = W[k*128+n]) so both operands load as float2.
__global__ void k_gemm_bias(const float* __restrict__ A, const float* __restrict__ Bt,
                            const float* __restrict__ bias, float* __restrict__ C, int N) {
  int wid  = (blockIdx.x * blockDim.x + threadIdx.x) >> 5;
  int lane = threadIdx.x & 31;
  int tilesM = (N + 15) >> 4;
  if (wid >= tilesM * (HIDC / 16)) return;
  int tm = wid >> 3;            // HID/16 == 8 column tiles
  int tn = wid & 7;
  int row0 = tm << 4, col0 = tn << 4;
  int m  = lane & 15;
  int kk = (lane >> 4) << 1;    // lanes 0-15: K pair {0,1}; lanes 16-31: {2,3}
  int ar = row0 + m; if (ar >= N) ar = N - 1;
  const float* Ap = A  + (long)ar * HIDC + kk;
  const float* Bp = Bt + (long)(col0 + m) * HIDC + kk;
  v8f acc = {};
#pragma unroll
  for (int k = 0; k < HIDC; k += 4) {
    v2f a = *(const v2f*)(Ap + k);
    v2f b = *(const v2f*)(Bp + k);
    acc = __builtin_amdgcn_wmma_f32_16x16x4_f32(false, a, false, b, (short)0, acc, false, false);
  }
  int n  = col0 + (lane & 15);
  int mb = row0 + ((lane >> 4) << 3);
  float bv = bias[n];
#pragma unroll
  for (int v = 0; v < 8; v++) {
    int r = mb + v;
    if (r < N) C[(long)r * HIDC + n] = acc[v] + bv;
  }
}

// ---------- semantic-attention GEMM: colsum[n] += sum_rows tanh((A@Wk)[r,n] + bk[n]) ----------
__global__ void k_group_colsum(const float* __restrict__ A, const float* __restrict__ Bt,
                               const float* __restrict__ bias, float* __restrict__ colsum, int N) {
  int wid  = (blockIdx.x * blockDim.x + threadIdx.x) >> 5;
  int lane = threadIdx.x & 31;
  int tilesM = (N + 15) >> 4;
  if (wid >= tilesM * (HIDC / 16)) return;
  int tm = wid >> 3, tn = wid & 7;
  int row0 = tm << 4, col0 = tn << 4;
  int m  = lane & 15;
  int kk = (lane >> 4) << 1;
  int ar = row0 + m; if (ar >= N) ar = N - 1;
  const float* Ap = A  + (long)ar * HIDC + kk;
  const float* Bp = Bt + (long)(col0 + m) * HIDC + kk;
  v8f acc = {};
#pragma unroll
  for (int k = 0; k < HIDC; k += 4) {
    v2f a = *(const v2f*)(Ap + k);
    v2f b = *(const v2f*)(Bp + k);
    acc = __builtin_amdgcn_wmma_f32_16x16x4_f32(false, a, false, b, (short)0, acc, false, false);
  }
  int n  = col0 + (lane & 15);
  int mb = row0 + ((lane >> 4) << 3);
  float bv = bias[n];
  float s = 0.f;
#pragma unroll
  for (int v = 0; v < 8; v++) {
    int r = mb + v;
    if (r < N) s += tanhf(acc[v] + bv);
  }
  s += __shfl_xor(s, 16, 32);       // combine the two lanes holding the same column
  if (lane < 16) atomicAdd(&colsum[n], s);
}

// ---------- per-node attention logits: out[n,h] = sum_d z[n,h,d]*avec[h,d] ----------
__global__ void k_alpha(const float* __restrict__ Z, const float* __restrict__ avec,
                        float* __restrict__ out, int N) {
  long i = (long)blockIdx.x * blockDim.x + threadIdx.x;
  if (i >= (long)N * HEADS) return;
  long n = i >> 3; int h = (int)(i & 7);
  const float4* z4 = (const float4*)(Z + n * HIDC + h * DH);
  const float4* a4 = (const float4*)(avec + h * DH);
  float s = 0.f;
#pragma unroll
  for (int j = 0; j < 4; j++) {
    float4 z = z4[j], a = a4[j];
    s += z.x * a.x + z.y * a.y + z.z * a.z + z.w * a.w;
  }
  out[i] = s;
}

// ---------- edge passes ----------
__global__ void k_edge_max(const int* __restrict__ src, const int* __restrict__ dst,
                           const float* __restrict__ asrc, const float* __restrict__ adst,
                           unsigned* __restrict__ dmax, long E8) {
  long i = (long)blockIdx.x * blockDim.x + threadIdx.x;
  if (i >= E8) return;
  long e = i >> 3; int h = (int)(i & 7);
  int s = src[e], d = dst[e];
  float al = asrc[(long)s * 8 + h] + adst[(long)d * 8 + h];
  al = al >= 0.f ? al : NEG_SLOPE * al;
  atomicMax(&dmax[(long)d * 8 + h], ord_enc(al));
}

__global__ void k_edge_expsum(const int* __restrict__ src, const int* __restrict__ dst,
                              const float* __restrict__ asrc, const float* __restrict__ adst,
                              const unsigned* __restrict__ dmax, float* __restrict__ denom,
                              float* __restrict__ exA, long E8) {
  long i = (long)blockIdx.x * blockDim.x + threadIdx.x;
  if (i >= E8) return;
  long e = i >> 3; int h = (int)(i & 7);
  int s = src[e], d = dst[e];
  float al = asrc[(long)s * 8 + h] + adst[(long)d * 8 + h];
  al = al >= 0.f ? al : NEG_SLOPE * al;
  float ex = __expf(al - ord_dec(dmax[(long)d * 8 + h]));
  exA[i] = ex;
  atomicAdd(&denom[(long)d * 8 + h], ex);
}

__global__ void k_edge_agg(const int* __restrict__ src, const int* __restrict__ dst,
                           const float* __restrict__ exA, const float* __restrict__ denom,
                           const float* __restrict__ Zsrc, float* __restrict__ out, long E8) {
  long i = (long)blockIdx.x * blockDim.x + threadIdx.x;
  if (i >= E8) return;
  long e = i >> 3; int h = (int)(i & 7);
  int s = src[e], d = dst[e];
  float w = exA[i] / (denom[(long)d * 8 + h] + 1e-16f);
  const float* zs = Zsrc + (long)s * HIDC + h * DH;
  float* od = out + (long)d * HIDC + h * DH;
#pragma unroll
  for (int j = 0; j < DH; j++) atomicAdd(od + j, zs[j] * w);
}

// ---------- semantic softmax over K=2 metapaths ----------
__global__ void k_score2(const float* __restrict__ cs0, const float* __restrict__ cs1,
                         const float* __restrict__ q, float invN, float* __restrict__ attn) {
  __shared__ float sh0[128], sh1[128];
  int f = threadIdx.x;
  sh0[f] = q[f] * cs0[f] * invN;
  sh1[f] = q[f] * cs1[f] * invN;
  __syncthreads();
  if (f == 0) {
    float s0 = 0.f, s1 = 0.f;
    for (int i = 0; i < 128; i++) { s0 += sh0[i]; s1 += sh1[i]; }
    float m = fmaxf(s0, s1);
    float e0 = __expf(s0 - m), e1 = __expf(s1 - m);
    float d = e0 + e1;
    attn[0] = e0 / d; attn[1] = e1 / d;
  }
}

__global__ void k_combine2(const float* __restrict__ A, const float* __restrict__ B,
                           const float* __restrict__ attn, float* __restrict__ out,
                           long n, int do_elu) {
  long i = (long)blockIdx.x * blockDim.x + threadIdx.x;
  if (i >= n) return;
  float v = attn[0] * A[i] + attn[1] * B[i];
  if (do_elu) v = v > 0.f ? v : expm1f(v);
  out[i] = v;
}

// ---------- final: out = elu(xp) @ Wout + bout ----------
__global__ void k_final(const float* __restrict__ X, const float* __restrict__ Wout,
                        const float* __restrict__ bout, float* __restrict__ out, int N) {
  long i = (long)blockIdx.x * blockDim.x + threadIdx.x;
  if (i >= (long)N * 8) return;
  long n = i >> 3; int oc = (int)(i & 7);
  const float* x = X + n * HIDC;
  float s = bout[oc];
  for (int k = 0; k < HIDC; k++) {
    float v = x[k];
    v = v > 0.f ? v : expm1f(v);
    s += v * Wout[k * 8 + oc];
  }
  out[i] = s;
}

// ====================== host side ======================
static inline unsigned grd(long n) { return (unsigned)((n + 255) / 256); }

static void launch_gemm(hipStream_t st, const float* A, const float* Bt, const float* bias,
                        float* C, int N) {
  long waves = (long)((N + 15) >> 4) * 8;
  long thr = waves * 32;
  k_gemm_bias<<<grd(thr), 256, 0, st>>>(A, Bt, bias, C, N);
}
static void launch_group(hipStream_t st, const float* A, const float* Bt, const float* bias,
                         float* colsum, int N) {
  long waves = (long)((N + 15) >> 4) * 8;
  long thr = waves * 32;
  k_group_colsum<<<grd(thr), 256, 0, st>>>(A, Bt, bias, colsum, N);
}

static void edge_type(hipStream_t st, const float* Zsrc, const float* Zdst,
                      const float* a_s, const float* a_d, const int* ei, long E,
                      int Nsrc, int Ndst, float* out,
                      float* asrc, float* adst, unsigned* dmax, float* denom, float* exA) {
  const int* src = ei;
  const int* dst = ei + E;
  long n8s = (long)Nsrc * 8, n8d = (long)Ndst * 8, E8 = E * 8;
  k_alpha<<<grd(n8s), 256, 0, st>>>(Zsrc, a_s, asrc, Nsrc);
  k_alpha<<<grd(n8d), 256, 0, st>>>(Zdst, a_d, adst, Ndst);
  k_fill_u32<<<grd(n8d), 256, 0, st>>>(dmax, n8d, ORD_NEG_INF);
  k_edge_max<<<grd(E8), 256, 0, st>>>(src, dst, asrc, adst, dmax, E8);
  k_zero_f32<<<grd(n8d), 256, 0, st>>>(denom, n8d);
  k_zero_f32<<<grd((long)Ndst * HIDC), 256, 0, st>>>(out, (long)Ndst * HIDC);
  k_edge_expsum<<<grd(E8), 256, 0, st>>>(src, dst, asrc, adst, dmax, denom, exA, E8);
  k_edge_agg<<<grd(E8), 256, 0, st>>>(src, dst, exA, denom, Zsrc, out, E8);
  k_relu<<<grd((long)Ndst * HIDC), 256, 0, st>>>(out, (long)Ndst * HIDC);
}

extern "C" void kernel_launch(void* const* d_in, const int* in_sizes, int n_in,
                              void* d_out, int out_size, void* d_ws, size_t ws_size,
                              hipStream_t stream) {
  const float* x_paper  = (const float*)d_in[0];
  const float* x_author = (const float*)d_in[1];
  const float* Wp_p = (const float*)d_in[2];
  const float* bp_p = (const float*)d_in[3];
  const float* Wp_a = (const float*)d_in[4];
  const float* bp_a = (const float*)d_in[5];
  const float* as_ap = (const float*)d_in[6];
  const float* ad_ap = (const float*)d_in[7];
  const float* as_pa = (const float*)d_in[8];
  const float* ad_pa = (const float*)d_in[9];
  const float* as_pp = (const float*)d_in[10];
  const float* ad_pp = (const float*)d_in[11];
  const float* Wk  = (const float*)d_in[12];
  const float* bk  = (const float*)d_in[13];
  const float* qv  = (const float*)d_in[14];
  const float* Wout = (const float*)d_in[15];
  const float* bout = (const float*)d_in[16];
  const int* ei_ap = (const int*)d_in[17];
  const int* ei_pa = (const int*)d_in[18];
  const int* ei_pp = (const int*)d_in[19];

  int NP = in_sizes[0] / HIDC;
  int NA = in_sizes[1] / HIDC;
  long E = in_sizes[17] / 2;
  int Nmax = NP > NA ? NP : NA;

  // workspace carve
  char* base = (char*)d_ws;
  size_t off = 0;
  auto carve = [&](size_t bytes) -> void* {
    void* p = base + off;
    off = (off + bytes + 255) & ~(size_t)255;
    return p;
  };
  float* zp      = (float*)carve((size_t)NP * HIDC * 4);
  float* za      = (float*)carve((size_t)NA * HIDC * 4);
  float* out_ap  = (float*)carve((size_t)NP * HIDC * 4);
  float* out_pp  = (float*)carve((size_t)NP * HIDC * 4);
  float* out_pa  = (float*)carve((size_t)NA * HIDC * 4);
  float* xp_buf  = (float*)carve((size_t)NP * HIDC * 4);
  float* asrc    = (float*)carve((size_t)Nmax * 8 * 4);
  float* adst    = (float*)carve((size_t)Nmax * 8 * 4);
  unsigned* dmax = (unsigned*)carve((size_t)Nmax * 8 * 4);
  float* denom   = (float*)carve((size_t)Nmax * 8 * 4);
  float* exA     = (float*)carve((size_t)E * 8 * 4);
  float* WtA     = (float*)carve(HIDC * HIDC * 4);
  float* WtK     = (float*)carve(HIDC * HIDC * 4);
  float* colsum  = (float*)carve(2 * HIDC * 4);   // [0..127]=ap, [128..255]=pp
  float* attn    = (float*)carve(2 * 4);

  const float* xp_cur = x_paper;
  const float* xa_cur = x_author;

  for (int l = 0; l < 2; l++) {
    const size_t wOff = (size_t)l * HIDC * HIDC;
    const size_t vOff = (size_t)l * HIDC;   // also == l*H*D for attention vectors

    // node-type projections (fp32 WMMA GEMMs)
    k_transpose128<<<64, 256, 0, stream>>>(Wp_p + wOff, WtA);
    launch_gemm(stream, xp_cur, WtA, bp_p + vOff, zp, NP);
    k_transpose128<<<64, 256, 0, stream>>>(Wp_a + wOff, WtA);
    launch_gemm(stream, xa_cur, WtA, bp_a + vOff, za, NA);

    // per-metapath GAT attention + aggregation
    edge_type(stream, za, zp, as_ap + vOff, ad_ap + vOff, ei_ap, E, NA, NP, out_ap,
              asrc, adst, dmax, denom, exA);
    edge_type(stream, zp, za, as_pa + vOff, ad_pa + vOff, ei_pa, E, NP, NA, out_pa,
              asrc, adst, dmax, denom, exA);
    edge_type(stream, zp, zp, as_pp + vOff, ad_pp + vOff, ei_pp, E, NP, NP, out_pp,
              asrc, adst, dmax, denom, exA);

    // semantic attention (paper: K=2). Author: softmax over 1 score == 1, skip.
    k_transpose128<<<64, 256, 0, stream>>>(Wk + wOff, WtK);
    k_zero_f32<<<1, 256, 0, stream>>>(colsum, 2 * HIDC);
    launch_group(stream, out_ap, WtK, bk + vOff, colsum, NP);
    launch_group(stream, out_pp, WtK, bk + vOff, colsum + HIDC, NP);
    k_score2<<<1, 128, 0, stream>>>(colsum, colsum + HIDC, qv + vOff, 1.f / (float)NP, attn);
    k_combine2<<<grd((long)NP * HIDC), 256, 0, stream>>>(out_ap, out_pp, attn, xp_buf,
                                                         (long)NP * HIDC, l < 1);
    if (l < 1)
      k_elu_inplace<<<grd((long)NA * HIDC), 256, 0, stream>>>(out_pa, (long)NA * HIDC);

    xp_cur = xp_buf;
    xa_cur = out_pa;
  }

  // final linear with fused ELU on the target node type
  k_final<<<grd((long)NP * 8), 256, 0, stream>>>(xp_cur, Wout, bout, (float*)d_out, NP);
}